// ContextualLoss_34626026340622
// MI455X (gfx1250) — compile-verified
//
#include <hip/hip_runtime.h>
#include <math.h>

typedef __attribute__((ext_vector_type(2))) float v2f;
typedef __attribute__((ext_vector_type(4))) float v4f;
typedef __attribute__((ext_vector_type(8))) float v8f;

#define NBATCH 4
#define CCH    64
#define PPIX   4096   // 64*64
#define EPSV   1e-5f

#define LDS_TILE_F   (16 * 68)            // padded 16-col x 64-k tile (floats)

// ---------------------------------------------------------------------------
// Kernel 1: per-channel mean of T over (N,H,W).  T is NCHW -> channel c of
// batch n is a contiguous 4096-float block.
// ---------------------------------------------------------------------------
__global__ void k_chan_mean(const float* __restrict__ T, float* __restrict__ meanT) {
    int c = blockIdx.x;            // 64 blocks
    float s = 0.f;
    for (int i = threadIdx.x; i < NBATCH * PPIX; i += 256) {
        int n = i >> 12, p = i & (PPIX - 1);
        s += T[((size_t)n * CCH + c) * PPIX + p];
    }
    __shared__ float sh[256];
    sh[threadIdx.x] = s; __syncthreads();
    for (int o = 128; o > 0; o >>= 1) {
        if (threadIdx.x < o) sh[threadIdx.x] += sh[threadIdx.x + o];
        __syncthreads();
    }
    if (threadIdx.x == 0) meanT[c] = sh[0] * (1.0f / (NBATCH * PPIX));
}

// ---------------------------------------------------------------------------
// Kernel 2: center by meanT, L2-normalize over C, transpose NCHW -> [n*P][C]
// (C contiguous — the layout both WMMA A and B fragments want).
// ---------------------------------------------------------------------------
__global__ void k_normalize(const float* __restrict__ T, const float* __restrict__ I,
                            const float* __restrict__ meanT,
                            float* __restrict__ Tn, float* __restrict__ In) {
    int idx = blockIdx.x * blockDim.x + threadIdx.x;   // 0..16383
    int n = idx >> 12, p = idx & (PPIX - 1);
    const float* tb = T + (size_t)n * CCH * PPIX + p;
    const float* ib = I + (size_t)n * CCH * PPIX + p;
    float st = 0.f, si = 0.f;
    for (int c = 0; c < CCH; ++c) {
        float m = meanT[c];
        float t = tb[(size_t)c * PPIX] - m;
        float i = ib[(size_t)c * PPIX] - m;
        st += t * t; si += i * i;
    }
    float rt = rsqrtf(st), ri = rsqrtf(si);
    float* to = Tn + (size_t)idx * CCH;
    float* io = In + (size_t)idx * CCH;
    for (int c = 0; c < CCH; ++c) {          // reloads hit L0/L2
        float m = meanT[c];
        to[c] = (tb[(size_t)c * PPIX] - m) * rt;
        io[c] = (ib[(size_t)c * PPIX] - m) * ri;
    }
}

// ---------------------------------------------------------------------------
// Kernel 3: zero kmax (harness poisons ws with 0xAA each run).
// ---------------------------------------------------------------------------
__global__ void k_zero(float* __restrict__ kmax) {
    kmax[blockIdx.x * 256 + threadIdx.x] = 0.f;
}

// ---------------------------------------------------------------------------
// Kernel 4: fused contextual kernel with async-LDS double buffering.
//   grid = 256 blocks x 128 threads (4 waves).  Block = (batch n, 64-row strip);
//   each wave owns 16 I-rows.  3 phases over all 256 column tiles:
//     A: row-min of raw = (1-cos)/2        (wave-local shfl reduction)
//     B: row-sum of exp(1 - raw/(min+eps)) (wave-local shfl reduction)
//     C: cs = w/rowsum; column-max via global atomicMax (positive-float int trick)
//   B tile (16 cols x 64 k, padded stride 68) streamed with
//   GLOBAL_LOAD_ASYNC_TO_LDS_B128 into a 2-deep LDS buffer: tile ct+1 is in
//   flight while the 16 WMMAs of tile ct execute; s_wait_asynccnt 0 + one
//   barrier per tile.
//   The LDS byte address is taken from the generic pointer's low 32 bits
//   (flat LDS addr = {aperture, lds_offset}; HW truncates to addr[31:0]) —
//   the ptrtoint also marks ldsB address-taken so the compiler cannot prove
//   the asm doesn't write it (round-2 bug: LDS reads were folded to undef).
//   The pointer is additionally passed as an unused asm operand to guarantee
//   escape.
// ---------------------------------------------------------------------------
__global__ __launch_bounds__(128)
void k_ctx_main(const float* __restrict__ In, const float* __restrict__ Tn,
                float* __restrict__ kmax) {
    __shared__ float ldsB[2][LDS_TILE_F];

    const int tid  = threadIdx.x;
    const int lane = tid & 31;
    const int wv   = tid >> 5;                       // wave in block: 0..3
    const int n    = blockIdx.x >> 6;                // 64 blocks per batch
    const int m0   = (blockIdx.x & 63) * 64 + wv * 16;

    // Preload A fragments for this wave's 16 rows (16 k-steps * v2f = 32 VGPRs).
    // f32 16x4 A layout: lane holds A[m = lane%16][k = j + 2*(lane>>4)], j=0..1.
    const float* arow = In + ((size_t)(n * PPIX + m0 + (lane & 15)) * CCH) + 2 * (lane >> 4);
    v2f a[16];
#pragma unroll
    for (int ks = 0; ks < 16; ++ks) a[ks] = *(const v2f*)(arow + ks * 4);

    const float* tbase = Tn + (size_t)n * PPIX * CCH;

    // Per-thread slice of the cooperative tile copy: 8 consecutive floats.
    const int    f0   = tid * 8;                        // 0..1016
    const float* gsrc = tbase + f0;                     // + ct*1024 per tile
    float* const dst0 = &ldsB[0][(f0 >> 6) * 68 + (f0 & 63)];
    float* const dst1 = &ldsB[1][(f0 >> 6) * 68 + (f0 & 63)];
    const uint32_t d0 = (uint32_t)(uintptr_t)dst0;      // LDS byte address
    const uint32_t d1 = (uint32_t)(uintptr_t)dst1;

    // issue async copy of tile ct into LDS buffer (ct & 1): two b128 per thread
    // (inst OFFSET is added to BOTH the global and LDS side, so one address
    //  register serves both 16-byte halves)
    auto issue_load = [&](int ct) {
        const float*   src = gsrc + (size_t)ct * 16 * CCH;
        const uint32_t d   = (ct & 1) ? d1 : d0;
        float* const   dp  = (ct & 1) ? dst1 : dst0;    // escape only
        asm volatile("global_load_async_to_lds_b128 %0, %1, off"
                     :: "v"(d), "v"(src), "v"(dp) : "memory");
        asm volatile("global_load_async_to_lds_b128 %0, %1, off offset:16"
                     :: "v"(d), "v"(src), "v"(dp) : "memory");
    };

    // 16 WMMA k-steps on the tile sitting in buffer (ct & 1)
    auto compute_tile = [&](int ct) -> v8f {
        const float* bp = &ldsB[ct & 1][(lane & 15) * 68 + 2 * (lane >> 4)];
        v8f acc = {};
#pragma unroll
        for (int ks = 0; ks < 16; ++ks) {
            v2f b = *(const v2f*)(bp + ks * 4);
            acc = __builtin_amdgcn_wmma_f32_16x16x4_f32(
                false, a[ks], false, b, (short)0, acc, false, false);
        }
        return acc;   // acc[r]: row (r + 8*(lane>>4)), col ct*16 + lane%16
    };

#define PHASE_PROLOGUE()                                       \
    issue_load(0);                                             \
    asm volatile("s_wait_asynccnt 0x0" ::: "memory");          \
    __syncthreads();

#define PHASE_EPILOGUE()                                       \
    asm volatile("s_wait_asynccnt 0x0" ::: "memory");          \
    __syncthreads();

    // ---- Phase A: per-row min of raw ----
    float rm[8];
#pragma unroll
    for (int r = 0; r < 8; ++r) rm[r] = 3.4e38f;
    PHASE_PROLOGUE();
    for (int ct = 0; ct < 256; ++ct) {
        issue_load((ct + 1) & 255);        // wrap: redundantly prefetches tile 0
        v8f acc = compute_tile(ct);
#pragma unroll
        for (int r = 0; r < 8; ++r) rm[r] = fminf(rm[r], 0.5f * (1.0f - acc[r]));
        PHASE_EPILOGUE();
    }
#pragma unroll
    for (int r = 0; r < 8; ++r) {
#pragma unroll
        for (int o = 1; o < 16; o <<= 1) rm[r] = fminf(rm[r], __shfl_xor(rm[r], o, 32));
        rm[r] = 1.0f / (rm[r] + EPSV);                // inverse denominator
    }

    // ---- Phase B: per-row sum of exp(1 - rel) ----
    float rs[8] = {0.f, 0.f, 0.f, 0.f, 0.f, 0.f, 0.f, 0.f};
    PHASE_PROLOGUE();
    for (int ct = 0; ct < 256; ++ct) {
        issue_load((ct + 1) & 255);
        v8f acc = compute_tile(ct);
#pragma unroll
        for (int r = 0; r < 8; ++r) {
            float raw = 0.5f * (1.0f - acc[r]);
            rs[r] += __expf(1.0f - raw * rm[r]);
        }
        PHASE_EPILOGUE();
    }
#pragma unroll
    for (int r = 0; r < 8; ++r) {
#pragma unroll
        for (int o = 1; o < 16; o <<= 1) rs[r] += __shfl_xor(rs[r], o, 32);
        rs[r] = 1.0f / rs[r];
    }

    // ---- Phase C: column max of cs, pushed to global kmax ----
    float* kb = kmax + (size_t)n * PPIX;
    PHASE_PROLOGUE();
    for (int ct = 0; ct < 256; ++ct) {
        issue_load((ct + 1) & 255);
        v8f acc = compute_tile(ct);
        float cmax = 0.f;
#pragma unroll
        for (int r = 0; r < 8; ++r) {
            float raw = 0.5f * (1.0f - acc[r]);
            cmax = fmaxf(cmax, __expf(1.0f - raw * rm[r]) * rs[r]);
        }
        cmax = fmaxf(cmax, __shfl_xor(cmax, 16, 32)); // merge row halves (same column)
        if (lane < 16)                                 // cs > 0 -> int order == float order
            atomicMax((int*)(kb + ct * 16 + lane), __float_as_int(cmax));
        PHASE_EPILOGUE();
    }

#undef PHASE_PROLOGUE
#undef PHASE_EPILOGUE
}

// ---------------------------------------------------------------------------
// Kernel 5: score = mean_n( -log( mean_p kmax[n][p] ) )
// ---------------------------------------------------------------------------
__global__ void k_final(const float* __restrict__ kmax, float* __restrict__ out) {
    __shared__ float sh[256];
    float score = 0.f;
    for (int n = 0; n < NBATCH; ++n) {
        float s = 0.f;
        for (int p = threadIdx.x; p < PPIX; p += 256) s += kmax[(size_t)n * PPIX + p];
        sh[threadIdx.x] = s; __syncthreads();
        for (int o = 128; o > 0; o >>= 1) {
            if (threadIdx.x < o) sh[threadIdx.x] += sh[threadIdx.x + o];
            __syncthreads();
        }
        if (threadIdx.x == 0) score += -logf(sh[0] * (1.0f / PPIX));
        __syncthreads();
    }
    if (threadIdx.x == 0) out[0] = score * (1.0f / NBATCH);
}

// ---------------------------------------------------------------------------
extern "C" void kernel_launch(void* const* d_in, const int* in_sizes, int n_in,
                              void* d_out, int out_size, void* d_ws, size_t ws_size,
                              hipStream_t stream) {
    const float* I = (const float*)d_in[0];   // I_features, NCHW f32
    const float* T = (const float*)d_in[1];   // T_features, NCHW f32

    float* ws    = (float*)d_ws;
    float* Tn    = ws;                                    // 4*4096*64 f32 (4 MB)
    float* In    = Tn + (size_t)NBATCH * PPIX * CCH;      // 4*4096*64 f32 (4 MB)
    float* meanT = In + (size_t)NBATCH * PPIX * CCH;      // 64 f32
    float* kmax  = meanT + CCH;                           // 4*4096 f32 (64 KB)

    k_chan_mean<<<CCH, 256, 0, stream>>>(T, meanT);
    k_normalize<<<(NBATCH * PPIX) / 256, 256, 0, stream>>>(T, I, meanT, Tn, In);
    k_zero<<<(NBATCH * PPIX) / 256, 256, 0, stream>>>(kmax);
    k_ctx_main<<<256, 128, 0, stream>>>(In, Tn, kmax);
    k_final<<<1, 256, 0, stream>>>(kmax, (float*)d_out);
}